// HamEvo_56006373540016
// MI455X (gfx1250) — compile-verified
//
#include <hip/hip_runtime.h>

typedef float v2f __attribute__((ext_vector_type(2)));
typedef float v8f __attribute__((ext_vector_type(8)));

#define BATCH   16
#define NSTEPS  10
#define GROUPS  (1 << 18)            // 2^18 independent 4-amplitude groups
#define STATE_ELEMS (GROUPS * 64)    // 2^20 * 16 floats per re/im plane
#define TILE    16                   // o-groups staged per LDS pass
#define OT      8                    // tiles per block
#define LPAD    66                   // padded LDS row stride (floats) -> bank-conflict free

// ---------- 4x4 complex matmul helper (setup kernel only) ----------
__device__ __forceinline__ void cmatmul4(const float* ar, const float* ai,
                                         const float* br, const float* bi,
                                         float* cr, float* ci) {
  for (int r = 0; r < 4; ++r)
    for (int c = 0; c < 4; ++c) {
      float sr = 0.f, si = 0.f;
      for (int k = 0; k < 4; ++k) {
        sr += ar[r*4+k]*br[k*4+c] - ai[r*4+k]*bi[k*4+c];
        si += ar[r*4+k]*bi[k*4+c] + ai[r*4+k]*br[k*4+c];
      }
      cr[r*4+c] = sr; ci[r*4+c] = si;
    }
}

// ---------- Setup: per-batch RK4 propagator P_b = R(hA)^10, emit WMMA A-fragments ----------
__global__ void hamevo_propagator(const float* __restrict__ H_re,
                                  const float* __restrict__ H_im,
                                  const float* __restrict__ t,
                                  float* __restrict__ afrag) {
  const int b = threadIdx.x;
  if (b >= BATCH) return;
  const int p[4] = {0, 2, 1, 3};     // bit-swap from pyqtorch index reversal + tensordot order
  const float h = t[b] / (float)NSTEPS;

  // X = h * (-i) * M,  M[r][c] = Hflat[p(r)][p(c)]
  float Xr[16], Xi[16];
  for (int r = 0; r < 4; ++r)
    for (int c = 0; c < 4; ++c) {
      const float mr = H_re[(p[r]*4 + p[c])*BATCH + b];
      const float mi = H_im[(p[r]*4 + p[c])*BATCH + b];
      Xr[r*4+c] =  h * mi;
      Xi[r*4+c] = -h * mr;
    }

  // R = I + X(I + (X/2)(I + (X/3)(I + X/4)))  == RK4 stability polynomial
  float Ur[16], Ui[16], Tr[16], Ti[16];
  for (int k = 0; k < 16; ++k) { Ur[k] = 0.25f*Xr[k]; Ui[k] = 0.25f*Xi[k]; }
  for (int d = 0; d < 4; ++d) Ur[d*5] += 1.f;
  const float invs[3] = {1.f/3.f, 0.5f, 1.f};
  for (int s = 0; s < 3; ++s) {
    cmatmul4(Xr, Xi, Ur, Ui, Tr, Ti);
    for (int k = 0; k < 16; ++k) { Ur[k] = invs[s]*Tr[k]; Ui[k] = invs[s]*Ti[k]; }
    for (int d = 0; d < 4; ++d) Ur[d*5] += 1.f;
  }

  // P = R^NSTEPS
  float Pr[16], Pi[16];
  for (int k = 0; k < 16; ++k) { Pr[k] = Ur[k]; Pi[k] = Ui[k]; }
  for (int s = 1; s < NSTEPS; ++s) {
    cmatmul4(Ur, Ui, Pr, Pi, Tr, Ti);
    for (int k = 0; k < 16; ++k) { Pr[k] = Tr[k]; Pi[k] = Ti[k]; }
  }

  // Real 8x8 propagator: [re;im]_out = [[Pr,-Pi],[Pi,Pr]] [re;im]_in
  float W[8][8];
  for (int r = 0; r < 4; ++r)
    for (int c = 0; c < 4; ++c) {
      W[r][c]     =  Pr[r*4+c];
      W[r][c+4]   = -Pi[r*4+c];
      W[r+4][c]   =  Pi[r*4+c];
      W[r+4][c+4] =  Pr[r*4+c];
    }

  // Emit A-matrix fragments for V_WMMA_F32_16X16X4_F32 (16x4, wave32):
  // lane l -> row m = l&15; VGPR v -> k_local = v + 2*(l>>4). Rows 8..15 zero-padded.
  // Fragment f selects K-columns [4f, 4f+4) of W.
  for (int f = 0; f < 2; ++f)
    for (int v = 0; v < 2; ++v)
      for (int l = 0; l < 32; ++l) {
        const int m = l & 15, kl = v + 2*(l >> 4);
        const float val = (m < 8) ? W[m][f*4 + kl] : 0.f;
        afrag[(((b*2 + f)*2 + v)*32) + l] = val;
      }
}

// ---------- Main: streamed application of P_b via WMMA, LDS-staged for coalescing ----------
__global__ void __launch_bounds__(512) hamevo_apply(const float* __restrict__ sre,
                                                    const float* __restrict__ sim,
                                                    const float* __restrict__ afrag,
                                                    float* __restrict__ out) {
  __shared__ float ls_re[TILE*LPAD], ls_im[TILE*LPAD];
  __shared__ float lo_re[TILE*LPAD], lo_im[TILE*LPAD];

  const int tid  = threadIdx.x;
  const int lane = tid & 31;
  const int b    = tid >> 5;               // wave id == batch index (16 waves)
  const int sn   = tid >> 5;               // staging row
  const int sx   = (tid & 31) * 2;         // staging col (float2)

  // Per-wave A fragments (persistent across all tiles)
  v2f a0, a1;
  a0.x = afrag[((b*2+0)*2+0)*32 + lane];
  a0.y = afrag[((b*2+0)*2+1)*32 + lane];
  a1.x = afrag[((b*2+1)*2+0)*32 + lane];
  a1.y = afrag[((b*2+1)*2+1)*32 + lane];

  // B-fragment LDS address: lane -> column n = lane&15, K-half jh = lane>>4, +16 per VGPR
  const int bn   = lane & 15;
  const int jh   = lane >> 4;
  const int boff = bn*LPAD + jh*32 + b;

  float* __restrict__ outre = out;
  float* __restrict__ outim = out + STATE_ELEMS;

  for (int it = 0; it < OT; ++it) {
    const long gbase = ((long)(blockIdx.x*OT + it)) * (TILE*64);

    // Coalesced stage-in: 512 threads x float2 x (re,im)
    const float2 r2 = *(const float2*)(sre + gbase + 2*tid);
    const float2 i2 = *(const float2*)(sim + gbase + 2*tid);
    *(float2*)&ls_re[sn*LPAD + sx] = r2;
    *(float2*)&ls_im[sn*LPAD + sx] = i2;
    __syncthreads();

    // Gather B fragments (conflict-free: stride LPAD=66 -> distinct banks for all 32 lanes)
    v2f br_, bi_;
    br_.x = ls_re[boff];      br_.y = ls_re[boff + 16];
    bi_.x = ls_im[boff];      bi_.y = ls_im[boff + 16];

    // D(8x16 useful) = W[:,0:4] @ S_re  +  W[:,4:8] @ S_im   (two chained 16x16x4 WMMAs)
    v8f acc = {};
    acc = __builtin_amdgcn_wmma_f32_16x16x4_f32(false, a0, false, br_, (short)0, acc, false, false);
    acc = __builtin_amdgcn_wmma_f32_16x16x4_f32(false, a1, false, bi_, (short)0, acc, false, false);

    // D rows 0..3 = out_re[j], rows 4..7 = out_im[j]; valid in lanes 0..15 (N = bn)
    if (lane < 16) {
      const int o = bn*LPAD + b;
      lo_re[o]      = acc[0]; lo_re[o + 16] = acc[1];
      lo_re[o + 32] = acc[2]; lo_re[o + 48] = acc[3];
      lo_im[o]      = acc[4]; lo_im[o + 16] = acc[5];
      lo_im[o + 32] = acc[6]; lo_im[o + 48] = acc[7];
    }
    __syncthreads();

    // Coalesced stage-out
    *(float2*)(outre + gbase + 2*tid) = *(const float2*)&lo_re[sn*LPAD + sx];
    *(float2*)(outim + gbase + 2*tid) = *(const float2*)&lo_im[sn*LPAD + sx];
    __syncthreads();
  }
}

extern "C" void kernel_launch(void* const* d_in, const int* in_sizes, int n_in,
                              void* d_out, int out_size, void* d_ws, size_t ws_size,
                              hipStream_t stream) {
  (void)in_sizes; (void)n_in; (void)out_size; (void)ws_size;
  const float* H_re = (const float*)d_in[0];
  const float* H_im = (const float*)d_in[1];
  const float* sre  = (const float*)d_in[2];
  const float* sim  = (const float*)d_in[3];
  const float* t    = (const float*)d_in[4];
  float* out   = (float*)d_out;
  float* afrag = (float*)d_ws;   // 16 batches * 128 floats = 8 KB of scratch

  hamevo_propagator<<<1, 32, 0, stream>>>(H_re, H_im, t, afrag);

  const int nblocks = GROUPS / (TILE * OT);  // 2048 blocks x 512 threads
  hamevo_apply<<<nblocks, 512, 0, stream>>>(sre, sim, afrag, out);
}